// PointTransformerEncoder_49520972923213
// MI455X (gfx1250) — compile-verified
//
#include <hip/hip_runtime.h>
#include <hip/hip_bf16.h>

typedef float v2f __attribute__((ext_vector_type(2)));
typedef float v8f __attribute__((ext_vector_type(8)));

// ---------------------------------------------------------------------------
// transpose: x (B,3,N) -> P (B,N,3)
// ---------------------------------------------------------------------------
__global__ void k_transpose(const float* __restrict__ x, float* __restrict__ P) {
  int gid = blockIdx.x * blockDim.x + threadIdx.x;
  if (gid >= 4 * 4096) return;
  int b = gid >> 12, i = gid & 4095;
  const float* xb = x + (size_t)b * 3 * 4096;
  float* pp = P + (size_t)gid * 3;
  pp[0] = xb[0 * 4096 + i];
  pp[1] = xb[1 * 4096 + i];
  pp[2] = xb[2 * 4096 + i];
}

// ---------------------------------------------------------------------------
// stage-0 transition down (stride 1): X0 = relu(bn(P0 @ W(3x32)))
// ---------------------------------------------------------------------------
__global__ void k_td0(const float* __restrict__ P, const float* __restrict__ W,
                      const float* __restrict__ bnb, const float* __restrict__ bns,
                      float* __restrict__ X) {
  int gid = blockIdx.x * blockDim.x + threadIdx.x;
  if (gid >= 4 * 4096 * 32) return;
  int c = gid & 31;
  int pidx = gid >> 5;
  const float* pp = P + (size_t)pidx * 3;
  float acc = pp[0] * W[0 * 32 + c] + pp[1] * W[1 * 32 + c] + pp[2] * W[2 * 32 + c];
  X[gid] = fmaxf(acc * bns[c] + bnb[c], 0.f);
}

// ---------------------------------------------------------------------------
// furthest point sampling: one workgroup per batch cloud.
// d[] lives in LDS; argmax via wave32 shuffles + LDS partials.
// ---------------------------------------------------------------------------
template <int NPTS, int M>
__global__ void k_fps(const float* __restrict__ P, int* __restrict__ sidx) {
  int b = blockIdx.x;
  const float* p = P + (size_t)b * NPTS * 3;
  int* out = sidx + (size_t)b * M;
  __shared__ float dmin[NPTS];
  __shared__ float pv[16];
  __shared__ int pi[16];
  __shared__ int curs;
  const int T = blockDim.x;  // 512 -> 16 waves
  int t = threadIdx.x;
  for (int k = t; k < NPTS; k += T) dmin[k] = 1e10f;
  if (t == 0) { out[0] = 0; curs = 0; }
  __syncthreads();
  for (int it = 1; it < M; ++it) {
    int cur = curs;
    float cx = p[cur * 3 + 0], cy = p[cur * 3 + 1], cz = p[cur * 3 + 2];
    float bv = -1.f; int bi = 0;
    for (int k = t; k < NPTS; k += T) {
      float dx = p[k * 3 + 0] - cx, dy = p[k * 3 + 1] - cy, dz = p[k * 3 + 2] - cz;
      float d = dx * dx + dy * dy + dz * dz;
      d = fminf(dmin[k], d);
      dmin[k] = d;
      if (d > bv) { bv = d; bi = k; }
    }
    // wave32 reduce (prefer larger d; tie -> lower index, matching argmax-first)
    for (int off = 16; off > 0; off >>= 1) {
      float ov = __shfl_down(bv, off, 32);
      int oi = __shfl_down(bi, off, 32);
      if (ov > bv || (ov == bv && oi < bi)) { bv = ov; bi = oi; }
    }
    int wid = t >> 5;
    if ((t & 31) == 0) { pv[wid] = bv; pi[wid] = bi; }
    __syncthreads();
    if (t == 0) {
      float fbv = pv[0]; int fbi = pi[0];
      int nw = T >> 5;
      for (int w = 1; w < nw; ++w)
        if (pv[w] > fbv || (pv[w] == fbv && pi[w] < fbi)) { fbv = pv[w]; fbi = pi[w]; }
      out[it] = fbi; curs = fbi;
    }
    __syncthreads();
  }
}

// gather selected coords: Pn[b][i] = P[b][sidx[b][i]]
__global__ void k_gather3(const float* __restrict__ P, const int* __restrict__ sidx,
                          float* __restrict__ Pn, int nold, int m) {
  int gid = blockIdx.x * blockDim.x + threadIdx.x;
  if (gid >= 4 * m) return;
  int b = gid / m, i = gid - b * m;
  int s = sidx[(size_t)b * m + i];
  const float* src = P + ((size_t)b * nold + s) * 3;
  float* dst = Pn + (size_t)gid * 3;
  dst[0] = src[0]; dst[1] = src[1]; dst[2] = src[2];
}

// ---------------------------------------------------------------------------
// kNN: thread-per-query, register-resident sorted top-K, LDS-staged src tiles.
// grid = (m/bs, B); Q/S are (B,m,3)/(B,nsrc,3).
// ---------------------------------------------------------------------------
template <int K>
__global__ void k_knn(const float* __restrict__ Q, const float* __restrict__ S,
                      int* __restrict__ idx, int m, int nsrc) {
  int b = blockIdx.y;
  int i = blockIdx.x * blockDim.x + threadIdx.x;
  const float* qp = Q + (size_t)b * m * 3;
  const float* sp = S + (size_t)b * nsrc * 3;
  __shared__ float tile[128 * 3];
  bool act = (i < m);
  float qx = 0.f, qy = 0.f, qz = 0.f;
  if (act) { qx = qp[i * 3]; qy = qp[i * 3 + 1]; qz = qp[i * 3 + 2]; }
  float bd[K]; int bi[K];
#pragma unroll
  for (int r = 0; r < K; ++r) { bd[r] = 3.4e38f; bi[r] = 0; }
  for (int t0 = 0; t0 < nsrc; t0 += 128) {
    int cnt = nsrc - t0; if (cnt > 128) cnt = 128;
    __syncthreads();
    for (int j = threadIdx.x; j < cnt * 3; j += blockDim.x) tile[j] = sp[t0 * 3 + j];
    __syncthreads();
    if (act) {
      for (int j = 0; j < cnt; ++j) {
        float dx = tile[j * 3] - qx, dy = tile[j * 3 + 1] - qy, dz = tile[j * 3 + 2] - qz;
        float d = dx * dx + dy * dy + dz * dz;
        if (d < bd[K - 1]) {
          bd[K - 1] = d; bi[K - 1] = t0 + j;
#pragma unroll
          for (int r = K - 1; r > 0; --r) {
            if (bd[r] < bd[r - 1]) {
              float td = bd[r]; bd[r] = bd[r - 1]; bd[r - 1] = td;
              int ti = bi[r]; bi[r] = bi[r - 1]; bi[r - 1] = ti;
            }
          }
        }
      }
    }
  }
  if (act) {
    int* op = idx + ((size_t)b * m + i) * K;
#pragma unroll
    for (int r = 0; r < K; ++r) op[r] = bi[r];
  }
}

// ---------------------------------------------------------------------------
// transition-down grouping: one block per sampled point; group matrix in LDS.
// x_new[i] = max_j relu(bn(concat(p[g]-np, x[g]) @ W))
// ---------------------------------------------------------------------------
template <int NSAMP, int FIN, int COUT>
__global__ void k_td_group(const float* __restrict__ Pold, const float* __restrict__ Xold,
                           const float* __restrict__ Pnew, const int* __restrict__ gidx,
                           const float* __restrict__ W, const float* __restrict__ bnb,
                           const float* __restrict__ bns, float* __restrict__ Xnew,
                           int nold, int m) {
  int b = blockIdx.y, i = blockIdx.x, t = threadIdx.x;
  const float* po = Pold + (size_t)b * nold * 3;
  const float* xo = Xold + (size_t)b * nold * (FIN - 3);
  const float* pn = Pnew + ((size_t)b * m + i) * 3;
  const int* gi = gidx + ((size_t)b * m + i) * NSAMP;
  __shared__ float g[NSAMP * FIN];
  __shared__ int gl[NSAMP];
  if (t < NSAMP) gl[t] = gi[t];
  __syncthreads();
  float nx = pn[0], ny = pn[1], nz = pn[2];
  for (int e = t; e < NSAMP * FIN; e += COUT) {
    int j = e / FIN, f = e - j * FIN;
    int gj = gl[j];
    float vv;
    if (f == 0)      vv = po[gj * 3 + 0] - nx;
    else if (f == 1) vv = po[gj * 3 + 1] - ny;
    else if (f == 2) vv = po[gj * 3 + 2] - nz;
    else             vv = xo[(size_t)gj * (FIN - 3) + (f - 3)];
    g[e] = vv;
  }
  __syncthreads();
  float sc = bns[t], bb = bnb[t];
  float best = 0.f;  // relu outputs are >= 0
  for (int j = 0; j < NSAMP; ++j) {
    float acc = 0.f;
    for (int f = 0; f < FIN; ++f) acc += g[j * FIN + f] * W[f * COUT + t];
    best = fmaxf(best, fmaxf(acc * sc + bb, 0.f));
  }
  Xnew[((size_t)b * m + i) * COUT + t] = best;
}

// ---------------------------------------------------------------------------
// fp32 WMMA GEMM: C(n x cout) = A(n x K) @ W(K x cout) with fused epilogues.
// One wave per 16x16 tile; K stepped by 4 via V_WMMA_F32_16X16X4_F32.
// Fragment layouts per CDNA5 ISA 7.12.2.
// ---------------------------------------------------------------------------
enum { EP_BIAS = 0, EP_BN_RELU = 1, EP_BN_RES_RELU = 2 };

template <int MODE>
__global__ void k_gemm16(const float* __restrict__ A, const float* __restrict__ W,
                         const float* __restrict__ scale, const float* __restrict__ bias,
                         const float* __restrict__ resid, float* __restrict__ C,
                         int n, int Kdim, int cout) {
  int bz = blockIdx.z;
  int m0 = blockIdx.x * 16, n0 = blockIdx.y * 16;
  const float* Ab = A + (size_t)bz * n * Kdim;
  const float* Rb = resid + (size_t)bz * n * cout;  // only deref'd in MODE 2
  float* Cb = C + (size_t)bz * n * cout;
  int lane = threadIdx.x;
  int r16 = lane & 15, hf = lane >> 4;
  v8f acc = {0.f, 0.f, 0.f, 0.f, 0.f, 0.f, 0.f, 0.f};
  for (int k0 = 0; k0 < Kdim; k0 += 4) {
    int ka = k0 + 2 * hf;
    v2f a, bf;
    a.x = Ab[(size_t)(m0 + r16) * Kdim + ka];
    a.y = Ab[(size_t)(m0 + r16) * Kdim + ka + 1];
    bf.x = W[(size_t)ka * cout + n0 + r16];
    bf.y = W[(size_t)(ka + 1) * cout + n0 + r16];
    acc = __builtin_amdgcn_wmma_f32_16x16x4_f32(false, a, false, bf, (short)0, acc,
                                                false, false);
  }
  int col = n0 + r16;
#pragma unroll
  for (int r = 0; r < 8; ++r) {
    int row = m0 + r + 8 * hf;
    float v = acc[r];
    if (MODE == EP_BIAS)            v += bias[col];
    else if (MODE == EP_BN_RELU)    v = fmaxf(v * scale[col] + bias[col], 0.f);
    else /* EP_BN_RES_RELU */       v = fmaxf(v * scale[col] + bias[col] +
                                              Rb[(size_t)row * cout + col], 0.f);
    Cb[(size_t)row * cout + col] = v;
  }
}

// ---------------------------------------------------------------------------
// fused vector attention: one block per query point, block = C threads.
// pr / w MLPs, softmax over neighbors, SHARE-grouped aggregation, bn2+relu.
// ---------------------------------------------------------------------------
template <int C, int CS, int NSAMP>
__global__ void k_attn(const float* __restrict__ P, const int* __restrict__ idx,
                       const float* __restrict__ Qb, const float* __restrict__ Kb,
                       const float* __restrict__ Vb,
                       const float* __restrict__ p1W, const float* __restrict__ p1b,
                       const float* __restrict__ pbnb, const float* __restrict__ pbns,
                       const float* __restrict__ p2W, const float* __restrict__ p2b,
                       const float* __restrict__ wbn0b, const float* __restrict__ wbn0s,
                       const float* __restrict__ waW, const float* __restrict__ wab,
                       const float* __restrict__ wbn1b, const float* __restrict__ wbn1s,
                       const float* __restrict__ wbW, const float* __restrict__ wbb,
                       const float* __restrict__ bn2b, const float* __restrict__ bn2s,
                       float* __restrict__ Y, int n) {
  extern __shared__ float sm[];
  float* pr = sm;                 // NSAMP*C
  float* tw = pr + NSAMP * C;     // NSAMP*C
  float* ww = tw + NSAMP * C;     // NSAMP*CS
  float* w2 = ww + NSAMP * CS;    // NSAMP*CS
  float* lp = w2 + NSAMP * CS;    // NSAMP*4
  int* nb = (int*)(lp + NSAMP * 4);

  int b = blockIdx.y, i = blockIdx.x, t = threadIdx.x;
  const float* pp = P + (size_t)b * n * 3;
  const float* q = Qb + ((size_t)b * n + i) * C;
  const float* Kc = Kb + (size_t)b * n * C;
  const float* Vc = Vb + (size_t)b * n * C;
  const int* id = idx + ((size_t)b * n + i) * NSAMP;

  if (t < NSAMP) {
    int gj = id[t];
    nb[t] = gj;
    float dx = pp[gj * 3 + 0] - pp[i * 3 + 0];
    float dy = pp[gj * 3 + 1] - pp[i * 3 + 1];
    float dz = pp[gj * 3 + 2] - pp[i * 3 + 2];
#pragma unroll
    for (int c2 = 0; c2 < 3; ++c2) {
      float l = dx * p1W[0 * 3 + c2] + dy * p1W[1 * 3 + c2] + dz * p1W[2 * 3 + c2] + p1b[c2];
      lp[t * 4 + c2] = fmaxf(l * pbns[c2] + pbnb[c2], 0.f);
    }
  }
  __syncthreads();
  float qv = q[t];
  float s0 = wbn0s[t], b0 = wbn0b[t];
  float pw0 = p2W[0 * C + t], pw1 = p2W[1 * C + t], pw2 = p2W[2 * C + t], pbb = p2b[t];
#pragma unroll
  for (int j = 0; j < NSAMP; ++j) {
    float prv = lp[j * 4 + 0] * pw0 + lp[j * 4 + 1] * pw1 + lp[j * 4 + 2] * pw2 + pbb;
    pr[j * C + t] = prv;
    float wv = Kc[(size_t)nb[j] * C + t] - qv + prv;
    tw[j * C + t] = fmaxf(wv * s0 + b0, 0.f);
  }
  __syncthreads();
  for (int task = t; task < NSAMP * CS; task += C) {
    int j = task / CS, u = task - j * CS;
    float acc = wab[u];
    for (int cc = 0; cc < C; ++cc) acc += tw[j * C + cc] * waW[cc * CS + u];
    ww[task] = fmaxf(acc * wbn1s[u] + wbn1b[u], 0.f);
  }
  __syncthreads();
  for (int task = t; task < NSAMP * CS; task += C) {
    int j = task / CS, u = task - j * CS;
    float acc = wbb[u];
    for (int uu = 0; uu < CS; ++uu) acc += ww[j * CS + uu] * wbW[uu * CS + u];
    w2[task] = acc;
  }
  __syncthreads();
  if (t < CS) {
    float mx = -3.4e38f;
    for (int j = 0; j < NSAMP; ++j) mx = fmaxf(mx, w2[j * CS + t]);
    float ssum = 0.f;
    for (int j = 0; j < NSAMP; ++j) { float e = __expf(w2[j * CS + t] - mx); w2[j * CS + t] = e; ssum += e; }
    float inv = 1.f / ssum;
    for (int j = 0; j < NSAMP; ++j) w2[j * CS + t] *= inv;
  }
  __syncthreads();
  int u = t & (CS - 1);
  float acc = 0.f;
#pragma unroll
  for (int j = 0; j < NSAMP; ++j)
    acc += (Vc[(size_t)nb[j] * C + t] + pr[j * C + t]) * w2[j * CS + u];
  Y[((size_t)b * n + i) * C + t] = fmaxf(acc * bn2s[t] + bn2b[t], 0.f);
}

// ---------------------------------------------------------------------------
// head: mean-pool(64) -> relu(bn(l1)) -> l2, one block per batch cloud
// ---------------------------------------------------------------------------
__global__ void k_head(const float* __restrict__ X3, const float* __restrict__ l1W,
                       const float* __restrict__ l1b, const float* __restrict__ bnb,
                       const float* __restrict__ bns, const float* __restrict__ l2W,
                       const float* __restrict__ l2b, float* __restrict__ out) {
  __shared__ float g[512];
  __shared__ float h[512];
  int b = blockIdx.x, t = threadIdx.x;
  const float* x = X3 + (size_t)b * 64 * 512;
  float acc = 0.f;
  for (int r = 0; r < 64; ++r) acc += x[r * 512 + t];
  g[t] = acc * (1.f / 64.f);
  __syncthreads();
  acc = l1b[t];
  for (int c = 0; c < 512; ++c) acc += g[c] * l1W[c * 512 + t];
  h[t] = fmaxf(acc * bns[t] + bnb[t], 0.f);
  __syncthreads();
  acc = l2b[t];
  for (int c = 0; c < 512; ++c) acc += h[c] * l2W[c * 512 + t];
  out[(size_t)b * 512 + t] = acc;
}

// ---------------------------------------------------------------------------
// host-side stage drivers
// ---------------------------------------------------------------------------
template <int Nn, int Cc, int CSs, int NSs>
static inline void run_block(void* const* d_in, int sb, const float* Ps, float* Xs,
                             float* T0, float* TQ, float* TK, float* TV, int* IDX,
                             hipStream_t stream) {
  auto F = [&](int i) { return (const float*)d_in[i]; };
  dim3 gg(Nn / 16, Cc / 16, 4);
  // h = relu(bn1(x @ W1))
  k_gemm16<EP_BN_RELU><<<gg, 32, 0, stream>>>(Xs, F(sb + 0), F(sb + 3), F(sb + 2), Xs, T0, Nn, Cc, Cc);
  constexpr int bs = (Nn < 128) ? Nn : 128;
  k_knn<NSs><<<dim3(Nn / bs, 4), bs, 0, stream>>>(Ps, Ps, IDX, Nn, Nn);
  // q / k / v
  k_gemm16<EP_BIAS><<<gg, 32, 0, stream>>>(T0, F(sb + 16), F(sb + 17), F(sb + 17), T0, TQ, Nn, Cc, Cc);
  k_gemm16<EP_BIAS><<<gg, 32, 0, stream>>>(T0, F(sb + 8),  F(sb + 9),  F(sb + 9),  T0, TK, Nn, Cc, Cc);
  k_gemm16<EP_BIAS><<<gg, 32, 0, stream>>>(T0, F(sb + 18), F(sb + 19), F(sb + 19), T0, TV, Nn, Cc, Cc);
  size_t shm = (size_t)(2 * NSs * Cc + 2 * NSs * CSs + NSs * 4) * sizeof(float) + NSs * sizeof(int);
  k_attn<Cc, CSs, NSs><<<dim3(Nn, 4), Cc, shm, stream>>>(
      Ps, IDX, TQ, TK, TV,
      F(sb + 10), F(sb + 11), F(sb + 14), F(sb + 15), F(sb + 12), F(sb + 13),
      F(sb + 24), F(sb + 25), F(sb + 20), F(sb + 21), F(sb + 26), F(sb + 27),
      F(sb + 22), F(sb + 23), F(sb + 4), F(sb + 5), T0, Nn);
  // x = relu(bn3(y @ W3) + x)
  k_gemm16<EP_BN_RES_RELU><<<gg, 32, 0, stream>>>(T0, F(sb + 1), F(sb + 7), F(sb + 6), Xs, Xs, Nn, Cc, Cc);
}

template <int Nold, int M, int NSs, int FIN, int COUT>
static inline void run_td(void* const* d_in, int sb, const float* Pold, const float* Xold,
                          float* Pnew, float* Xnew, int* SIDX, int* GIDX,
                          hipStream_t stream) {
  auto F = [&](int i) { return (const float*)d_in[i]; };
  k_fps<Nold, M><<<4, 512, 0, stream>>>(Pold, SIDX);
  k_gather3<<<(4 * M + 127) / 128, 128, 0, stream>>>(Pold, SIDX, Pnew, Nold, M);
  constexpr int bs = (M < 128) ? M : 128;
  k_knn<NSs><<<dim3(M / bs, 4), bs, 0, stream>>>(Pnew, Pold, GIDX, M, Nold);
  k_td_group<NSs, FIN, COUT><<<dim3(M, 4), COUT, 0, stream>>>(
      Pold, Xold, Pnew, GIDX, F(sb + 28), F(sb + 29), F(sb + 30), Xnew, Nold, M);
}

extern "C" void kernel_launch(void* const* d_in, const int* in_sizes, int n_in,
                              void* d_out, int out_size, void* d_ws, size_t ws_size,
                              hipStream_t stream) {
  (void)in_sizes; (void)n_in; (void)out_size; (void)ws_size;
  // pytree (sorted-key) flatten order:
  //   0..5   head: bn.b, bn.s, l1.W, l1.b, l2.W, l2.b
  //   6+31*s stage s: blk{W1,W3,bn1.b,bn1.s,bn2.b,bn2.s,bn3.b,bn3.s,
  //                       k.W,k.b,p1.W,p1.b,p2.W,p2.b,pbn.b,pbn.s,q.W,q.b,v.W,v.b,
  //                       wa.W,wa.b,wb.W,wb.b,wbn0.b,wbn0.s,wbn1.b,wbn1.s},
  //                   td{W,bn.b,bn.s}
  //   130    x_bcn (4,3,4096)
  const float* xb = (const float*)d_in[130];
  float* ws = (float*)d_ws;
  float* P0 = ws;                  // 4*4096*3
  float* X0 = P0 + 49152;          // 4*4096*32
  float* P1 = X0 + 524288;         // 4*1024*3
  float* X1 = P1 + 12288;          // 4*1024*64
  float* P2 = X1 + 262144;         // 4*256*3
  float* X2 = P2 + 3072;           // 4*256*128
  float* P3 = X2 + 131072;         // 4*64*3
  float* X3 = P3 + 768;            // 4*64*512
  float* T0 = X3 + 131072;         // 4*4096*32 scratch (h / y)
  float* TQ = T0 + 524288;
  float* TK = TQ + 524288;
  float* TV = TK + 524288;
  int* IDX  = (int*)(TV + 524288); // 4*4096*16
  int* SIDX = IDX + 262144;        // 4*1024
  int* GIDX = SIDX + 4096;         // 4*1024*16

  k_transpose<<<64, 256, 0, stream>>>(xb, P0);
  k_td0<<<2048, 256, 0, stream>>>(P0, (const float*)d_in[6 + 28],
                                  (const float*)d_in[6 + 29], (const float*)d_in[6 + 30], X0);
  run_block<4096, 32, 4, 8>(d_in, 6, P0, X0, T0, TQ, TK, TV, IDX, stream);

  run_td<4096, 1024, 16, 35, 64>(d_in, 37, P0, X0, P1, X1, SIDX, GIDX, stream);
  run_block<1024, 64, 8, 16>(d_in, 37, P1, X1, T0, TQ, TK, TV, IDX, stream);

  run_td<1024, 256, 16, 67, 128>(d_in, 68, P1, X1, P2, X2, SIDX, GIDX, stream);
  run_block<256, 128, 16, 16>(d_in, 68, P2, X2, T0, TQ, TK, TV, IDX, stream);

  run_td<256, 64, 16, 131, 512>(d_in, 99, P2, X2, P3, X3, SIDX, GIDX, stream);
  run_block<64, 512, 64, 16>(d_in, 99, P3, X3, T0, TQ, TK, TV, IDX, stream);

  k_head<<<4, 512, 0, stream>>>(X3, (const float*)d_in[2], (const float*)d_in[3],
                                (const float*)d_in[0], (const float*)d_in[1],
                                (const float*)d_in[4], (const float*)d_in[5],
                                (float*)d_out);
}